// MultiBoxLoss_44959717654927
// MI455X (gfx1250) — compile-verified
//
#include <hip/hip_runtime.h>
#include <math.h>

#define BATCH      128
#define N_PRIORS   8732
#define N_OBJ      32
#define N_CLASSES  21
#define THRESH     0.5f
#define NEG_RATIO  3
#define ALPHA_C    10.0f
#define CHUNK      256
#define NCHUNKS    ((N_PRIORS + CHUNK - 1) / CHUNK)   // 35

// ---- gfx1250 async global->LDS staging (guarded; falls back to plain LDS stores) ----
#if defined(__HIP_DEVICE_COMPILE__)
#if __has_builtin(__builtin_amdgcn_global_load_async_to_lds_b128) && \
    __has_builtin(__builtin_amdgcn_s_wait_asynccnt)
#define ASYNC_LDS 1
typedef int v4i __attribute__((vector_size(16)));
typedef __attribute__((address_space(1))) v4i gv4i;   // global (AS1) int4
typedef __attribute__((address_space(3))) v4i lv4i;   // LDS    (AS3) int4
#endif
#endif

// ---------------------------------------------------------------------------
// Kernel 1: per object, argmax over priors of IoU (ties -> lowest prior idx)
// ---------------------------------------------------------------------------
__global__ __launch_bounds__(256)
void k_best_prior(const float* __restrict__ b_boxes,   // (B,M,4) xyxy
                  const float* __restrict__ priors,    // (P,4) cxcywh
                  int* __restrict__ best_prior)        // (B,M)
{
    const int b   = blockIdx.x;
    const int tid = threadIdx.x;
    __shared__ float sx1[N_OBJ], sy1[N_OBJ], sx2[N_OBJ], sy2[N_OBJ], sar[N_OBJ];
    __shared__ float rv[256];
    __shared__ int   ri[256];

    if (tid < N_OBJ) {
        const float4 bb = ((const float4*)b_boxes)[b * N_OBJ + tid];
        sx1[tid] = bb.x; sy1[tid] = bb.y; sx2[tid] = bb.z; sy2[tid] = bb.w;
        sar[tid] = (bb.z - bb.x) * (bb.w - bb.y);
    }
    __syncthreads();

    float bestV[N_OBJ];
    int   bestI[N_OBJ];
#pragma unroll
    for (int m = 0; m < N_OBJ; ++m) { bestV[m] = -1.0f; bestI[m] = 0; }

    for (int p = tid; p < N_PRIORS; p += 256) {
        const float4 pr = ((const float4*)priors)[p];
        const float px1 = pr.x - pr.z * 0.5f, py1 = pr.y - pr.w * 0.5f;
        const float px2 = pr.x + pr.z * 0.5f, py2 = pr.y + pr.w * 0.5f;
        const float parea = pr.z * pr.w;
#pragma unroll
        for (int m = 0; m < N_OBJ; ++m) {
            float iw = fminf(sx2[m], px2) - fmaxf(sx1[m], px1);
            float ih = fminf(sy2[m], py2) - fmaxf(sy1[m], py1);
            iw = fmaxf(iw, 0.0f); ih = fmaxf(ih, 0.0f);
            const float inter = iw * ih;
            const float iou = inter / (sar[m] + parea - inter);
            if (iou > bestV[m]) { bestV[m] = iou; bestI[m] = p; }  // strict > keeps first
        }
    }

    for (int m = 0; m < N_OBJ; ++m) {
        rv[tid] = bestV[m]; ri[tid] = bestI[m];
        __syncthreads();
        for (int s = 128; s > 0; s >>= 1) {
            if (tid < s) {
                if (rv[tid + s] > rv[tid] ||
                    (rv[tid + s] == rv[tid] && ri[tid + s] < ri[tid])) {
                    rv[tid] = rv[tid + s]; ri[tid] = ri[tid + s];
                }
            }
            __syncthreads();
        }
        if (tid == 0) best_prior[b * N_OBJ + m] = ri[0];
        __syncthreads();
    }
}

// ---------------------------------------------------------------------------
// Kernel 2: match priors, loc-L1 partials, CE via log-softmax, conf_neg out.
// pred_cls tile staged coalesced into LDS (async b128 on gfx1250).
// ---------------------------------------------------------------------------
__global__ __launch_bounds__(256)
void k_match_ce(const float* __restrict__ pred_loc,    // (B,P,4)
                const float* __restrict__ pred_cls,    // (B,P,21)
                const float* __restrict__ b_boxes,     // (B,M,4)
                const int*   __restrict__ b_labels,    // (B,M)
                const float* __restrict__ priors,      // (P,4)
                const int*   __restrict__ best_prior,  // (B,M)
                float* __restrict__ conf_neg,          // (B,P)
                float* __restrict__ partial_loc,       // (B*NCHUNKS)
                float* __restrict__ partial_conf,      // (B*NCHUNKS)
                int*   __restrict__ n_pos)             // (B)
{
    const int b   = blockIdx.y;
    const int p0  = blockIdx.x * CHUNK;
    const int tid = threadIdx.x;

    __shared__ float4 scls4[CHUNK * N_CLASSES / 4];    // 256*21 floats = 21.5 KB
    __shared__ float sx1[N_OBJ], sy1[N_OBJ], sx2[N_OBJ], sy2[N_OBJ], sar[N_OBJ];
    __shared__ int   slab[N_OBJ], sbp[N_OBJ];
    __shared__ float red[256];
    __shared__ int   ired[256];
    float* scls = (float*)scls4;

    if (tid < N_OBJ) {
        const float4 bb = ((const float4*)b_boxes)[b * N_OBJ + tid];
        sx1[tid] = bb.x; sy1[tid] = bb.y; sx2[tid] = bb.z; sy2[tid] = bb.w;
        sar[tid] = (bb.z - bb.x) * (bb.w - bb.y);
        slab[tid] = b_labels[b * N_OBJ + tid];
        sbp[tid]  = best_prior[b * N_OBJ + tid];
    }

    const int n   = min(CHUNK, N_PRIORS - p0);
    const int nf4 = n * N_CLASSES / 4;                 // chunk sizes 256 & 28 -> divisible by 4
    const float4* src4 =
        (const float4*)(pred_cls + ((size_t)b * N_PRIORS + p0) * N_CLASSES);

    const int p = p0 + tid;
    if (tid < n)  // prefetch hint for the loc row we'll need after staging
        __builtin_prefetch(pred_loc + ((size_t)b * N_PRIORS + p) * 4, 0, 0);

#ifdef ASYNC_LDS
    for (int i = tid; i < nf4; i += 256)
        __builtin_amdgcn_global_load_async_to_lds_b128(
            (gv4i*)(src4 + i), (lv4i*)(scls4 + i), 0, 0);
    __builtin_amdgcn_s_wait_asynccnt(0);
#else
    for (int i = tid; i < nf4; i += 256) scls4[i] = src4[i];
#endif
    __syncthreads();

    float locp = 0.0f, confp = 0.0f;
    int   posc = 0;
    if (tid < n) {
        const float4 pr = ((const float4*)priors)[p];
        const float px1 = pr.x - pr.z * 0.5f, py1 = pr.y - pr.w * 0.5f;
        const float px2 = pr.x + pr.z * 0.5f, py2 = pr.y + pr.w * 0.5f;
        const float parea = pr.z * pr.w;

        float best = -1.0f; int bm = 0;
#pragma unroll
        for (int m = 0; m < N_OBJ; ++m) {
            float iw = fminf(sx2[m], px2) - fmaxf(sx1[m], px1);
            float ih = fminf(sy2[m], py2) - fmaxf(sy1[m], py1);
            iw = fmaxf(iw, 0.0f); ih = fmaxf(ih, 0.0f);
            const float inter = iw * ih;
            float iou = inter / (sar[m] + parea - inter);
            if (p == sbp[m]) iou = 1.0f;               // best-prior-per-object override
            if (iou > best) { best = iou; bm = m; }    // first-max semantics
        }
        const bool pos = best >= THRESH;

        // cross-entropy with log-softmax over 21 classes (from LDS)
        const float* row = scls + tid * N_CLASSES;
        float mx = row[0];
#pragma unroll
        for (int c = 1; c < N_CLASSES; ++c) mx = fmaxf(mx, row[c]);
        float se = 0.0f;
#pragma unroll
        for (int c = 0; c < N_CLASSES; ++c) se += expf(row[c] - mx);
        const float lse = mx + logf(se);
        const int tc = pos ? slab[bm] : 0;
        const float ce = lse - row[tc];

        conf_neg[(size_t)b * N_PRIORS + p] = pos ? 0.0f : ce;

        if (pos) {
            confp = ce; posc = 1;
            const float x1 = sx1[bm], y1 = sy1[bm], x2 = sx2[bm], y2 = sy2[bm];
            const float cx = (x1 + x2) * 0.5f, cy = (y1 + y2) * 0.5f;
            const float w = x2 - x1, h = y2 - y1;
            const float g0 = (cx - pr.x) / (pr.z / 10.0f);
            const float g1 = (cy - pr.y) / (pr.w / 10.0f);
            const float g2 = logf(w / pr.z) * 5.0f;
            const float g3 = logf(h / pr.w) * 5.0f;
            const float4 pl = ((const float4*)pred_loc)[(size_t)b * N_PRIORS + p];
            locp = fabsf(pl.x - g0) + fabsf(pl.y - g1) +
                   fabsf(pl.z - g2) + fabsf(pl.w - g3);
        }
    }

    // deterministic block reductions
    float locT = 0.0f, confT = 0.0f;
    red[tid] = locp; __syncthreads();
    for (int s = 128; s > 0; s >>= 1) { if (tid < s) red[tid] += red[tid + s]; __syncthreads(); }
    if (tid == 0) locT = red[0];
    __syncthreads();
    red[tid] = confp; __syncthreads();
    for (int s = 128; s > 0; s >>= 1) { if (tid < s) red[tid] += red[tid + s]; __syncthreads(); }
    if (tid == 0) confT = red[0];
    __syncthreads();
    ired[tid] = posc; __syncthreads();
    for (int s = 128; s > 0; s >>= 1) { if (tid < s) ired[tid] += ired[tid + s]; __syncthreads(); }

    if (tid == 0) {
        const int blk = b * NCHUNKS + blockIdx.x;
        partial_loc[blk]  = locT;
        partial_conf[blk] = confT;
        atomicAdd(&n_pos[b], ired[0]);   // integer atomic: deterministic
    }
}

// ---------------------------------------------------------------------------
// Kernel 3: sum of top-K conf_neg per image (K = 3*n_pos) via bit-monotonic
// binary search for the K-th largest (all values >= 0). Row lives in LDS.
// ---------------------------------------------------------------------------
__global__ __launch_bounds__(256)
void k_hardneg(const float* __restrict__ conf_neg,
               const int*   __restrict__ n_pos,
               float* __restrict__ hardneg)
{
    const int b   = blockIdx.x;
    const int tid = threadIdx.x;
    __shared__ float sneg[N_PRIORS];      // 34.9 KB (<< 320 KB WGP LDS)
    __shared__ int   ired[256];
    __shared__ float fred[256];

    for (int i = tid; i < N_PRIORS; i += 256)
        sneg[i] = conf_neg[(size_t)b * N_PRIORS + i];
    __syncthreads();

    const int K = min(NEG_RATIO * n_pos[b], N_PRIORS);
    if (K <= 0) { if (tid == 0) hardneg[b] = 0.0f; return; }

    unsigned lo = 0u, hi = 0x7f7fffffu;
    while (lo < hi) {
        const unsigned mid = lo + (hi - lo + 1u) / 2u;
        const float t = __uint_as_float(mid);
        int c = 0;
        for (int i = tid; i < N_PRIORS; i += 256) c += (sneg[i] >= t) ? 1 : 0;
        ired[tid] = c; __syncthreads();
        for (int s = 128; s > 0; s >>= 1) { if (tid < s) ired[tid] += ired[tid + s]; __syncthreads(); }
        const int total = ired[0];
        __syncthreads();
        if (total >= K) lo = mid; else hi = mid - 1u;
    }
    const float t = __uint_as_float(lo);   // K-th largest value

    int g = 0; float s = 0.0f;
    for (int i = tid; i < N_PRIORS; i += 256) {
        const float v = sneg[i];
        if (v > t) { ++g; s += v; }
    }
    ired[tid] = g; fred[tid] = s; __syncthreads();
    for (int st = 128; st > 0; st >>= 1) {
        if (tid < st) { ired[tid] += ired[tid + st]; fred[tid] += fred[tid + st]; }
        __syncthreads();
    }
    if (tid == 0) hardneg[b] = fred[0] + (float)(K - ired[0]) * t;
}

// ---------------------------------------------------------------------------
// Kernel 4: fixed-order final reduction -> (total, loc_loss, conf_loss)
// ---------------------------------------------------------------------------
__global__ __launch_bounds__(256)
void k_final(const float* __restrict__ partial_loc,
             const float* __restrict__ partial_conf,
             const float* __restrict__ hardneg,
             const int*   __restrict__ n_pos,
             float* __restrict__ out)
{
    const int tid = threadIdx.x;
    __shared__ float r0[256], r1[256], r2[256];
    __shared__ int   r3[256];
    float a = 0.0f, bsum = 0.0f, c = 0.0f; int d = 0;
    for (int i = tid; i < BATCH * NCHUNKS; i += 256) { a += partial_loc[i]; bsum += partial_conf[i]; }
    for (int i = tid; i < BATCH; i += 256)           { c += hardneg[i]; d += n_pos[i]; }
    r0[tid] = a; r1[tid] = bsum; r2[tid] = c; r3[tid] = d;
    __syncthreads();
    for (int s = 128; s > 0; s >>= 1) {
        if (tid < s) { r0[tid] += r0[tid+s]; r1[tid] += r1[tid+s]; r2[tid] += r2[tid+s]; r3[tid] += r3[tid+s]; }
        __syncthreads();
    }
    if (tid == 0) {
        const float N = (float)r3[0];
        const float loc_loss  = ALPHA_C * (r0[0] / (N * 4.0f));
        const float conf_loss = (r2[0] + r1[0]) / N;
        out[0] = conf_loss + loc_loss;
        out[1] = loc_loss;
        out[2] = conf_loss;
    }
}

__global__ void k_init(int* __restrict__ n_pos)
{
    if (threadIdx.x < BATCH) n_pos[threadIdx.x] = 0;
}

// ---------------------------------------------------------------------------
extern "C" void kernel_launch(void* const* d_in, const int* in_sizes, int n_in,
                              void* d_out, int out_size, void* d_ws, size_t ws_size,
                              hipStream_t stream)
{
    (void)in_sizes; (void)n_in; (void)out_size; (void)ws_size;
    const float* pred_loc = (const float*)d_in[0];
    const float* pred_cls = (const float*)d_in[1];
    const float* b_boxes  = (const float*)d_in[2];
    const int*   b_labels = (const int*)d_in[3];
    const float* priors   = (const float*)d_in[4];
    float* out = (float*)d_out;

    // workspace layout (~4.6 MB total)
    int*   npos  = (int*)d_ws;                       // B ints
    int*   bestp = npos + BATCH;                     // B*M ints
    float* pl    = (float*)(bestp + BATCH * N_OBJ);  // B*NCHUNKS floats
    float* pc    = pl + BATCH * NCHUNKS;             // B*NCHUNKS floats
    float* hn    = pc + BATCH * NCHUNKS;             // B floats
    float* cneg  = hn + BATCH;                       // B*P floats

    k_init<<<1, 128, 0, stream>>>(npos);
    k_best_prior<<<BATCH, 256, 0, stream>>>(b_boxes, priors, bestp);
    k_match_ce<<<dim3(NCHUNKS, BATCH), 256, 0, stream>>>(
        pred_loc, pred_cls, b_boxes, b_labels, priors, bestp, cneg, pl, pc, npos);
    k_hardneg<<<BATCH, 256, 0, stream>>>(cneg, npos, hn);
    k_final<<<1, 256, 0, stream>>>(pl, pc, hn, npos, out);
}